// ConfusionMatrix_24532853194784
// MI455X (gfx1250) — compile-verified
//
#include <hip/hip_runtime.h>
#include <stdint.h>

// Confusion-matrix histogram for MI455X (gfx1250).
//
// Roofline: ~64 MiB of int32 index reads @ 23.3 TB/s => ~2.8 us floor; 0 FLOPs,
// so matrix cores are irrelevant — the win comes from CDNA5 data movement:
//  * Whole 150x150 u32 histogram (90 KB) privatized in LDS per workgroup
//    (enabled by CDNA5's 320 KB/WGP LDS). ds_add_u32 atomics, zero global
//    contention during streaming.
//  * Index streams staged global->LDS via the CDNA5 async-copy path
//    (global_load_async_to_lds_b128 / ASYNCcnt), 4-deep buffered with
//    s_wait_asynccnt. One 1024-thread block (32 waves) per WGP keeps
//    ~96 KB of async traffic in flight per WGP — above the per-WGP
//    bandwidth-delay product (~150 GB/s x ~0.6us), so HBM stays saturated.
//  * 128 private histograms merged with global_atomic_add_f32 (2.88M atomics
//    into a 90 KB L2-resident region; 128 colliding adds per address).

#define NUM_CLASSES 150
#define BINS (NUM_CLASSES * NUM_CLASSES)      // 22500
#define BLOCK 1024                            // 32 waves = full WGP
#define VEC 4                                 // int4 per thread per chunk
#define CHUNK (BLOCK * VEC)                   // 4096 elements per block-iter
#define STAGE_BYTES (BLOCK * VEC * 4)         // 16384 B per array per buffer
#define BUF_BYTES (2 * STAGE_BYTES)           // pred + targ: 32768 B / buffer
#define NBUF 4                                // async pipeline depth
#define GRID 128                              // ~one block per WGP

__global__ void cm_init_kernel(const float* __restrict__ cm,
                               float* __restrict__ out, int n) {
    int i = blockIdx.x * blockDim.x + threadIdx.x;
    if (i < n) out[i] = cm[i];
}

__global__ __launch_bounds__(BLOCK)
void cm_hist_kernel(const int* __restrict__ pred,
                    const int* __restrict__ targ,
                    float* __restrict__ out, int n) {
    // 90000 B histogram + 4 x 32768 B stage = ~216 KB static LDS (< 320 KB).
    __shared__ unsigned int hist[BINS];
    __shared__ __align__(16) unsigned char stage[NBUF * BUF_BYTES];

    const int tid = threadIdx.x;

    for (int i = tid; i < BINS; i += BLOCK) hist[i] = 0u;
    __syncthreads();

    const long long nChunks = (long long)n / CHUNK;
    const long long stride  = gridDim.x;
    const long long myChunks =
        (nChunks > (long long)blockIdx.x)
            ? ((nChunks - 1 - blockIdx.x) / stride + 1)
            : 0;

    // Issue one chunk's async copies (pred 16B/lane + targ 16B/lane) into
    // staging buffer `buf`. LDS dest address operand is the group-segment
    // byte offset (low 32 bits of the generic shared pointer).
    auto issue = [&](long long i, int buf) {
        long long chunk = (long long)blockIdx.x + i * stride;
        long long base  = chunk * CHUNK + (long long)tid * VEC;
        const int* gp = pred + base;
        const int* gt = targ + base;
        unsigned int lp =
            (unsigned int)(uintptr_t)(&stage[buf * BUF_BYTES + tid * 16]);
        unsigned int lt = lp + STAGE_BYTES;
        asm volatile("global_load_async_to_lds_b128 %0, %1, off"
                     :: "v"(lp), "v"(gp) : "memory");
        asm volatile("global_load_async_to_lds_b128 %0, %1, off"
                     :: "v"(lt), "v"(gt) : "memory");
    };

    // Prologue: fill the pipeline NBUF deep.
    for (long long i = 0; i < NBUF && i < myChunks; ++i)
        issue(i, (int)i);

    for (long long i = 0; i < myChunks; ++i) {
        int buf = (int)(i % NBUF);
        if (i + NBUF <= myChunks) {
            // 2*(NBUF-1) newer ops may stay outstanding; in-order completion
            // guarantees this buffer's 2 loads have landed.
            asm volatile("s_wait_asynccnt 0x6" ::: "memory");
        } else {
            asm volatile("s_wait_asynccnt 0x0" ::: "memory");
        }

        const int4* sp =
            (const int4*)(&stage[buf * BUF_BYTES + tid * 16]);
        const int4* st =
            (const int4*)(&stage[buf * BUF_BYTES + STAGE_BYTES + tid * 16]);
        int4 p = *sp;   // ds_load_b128
        int4 t = *st;   // ds_load_b128

        atomicAdd(&hist[t.x * NUM_CLASSES + p.x], 1u);  // ds_add_u32
        atomicAdd(&hist[t.y * NUM_CLASSES + p.y], 1u);
        atomicAdd(&hist[t.z * NUM_CLASSES + p.z], 1u);
        atomicAdd(&hist[t.w * NUM_CLASSES + p.w], 1u);

        // Refill this buffer for iteration i+NBUF. Its data was just consumed:
        // the ds_loads' operand dependency drained DScnt before the atomics,
        // and asm volatile ordering keeps this issue after those reads.
        if (i + NBUF < myChunks) issue(i + NBUF, buf);
    }

    // Tail (n % CHUNK) — never taken for N = 8388608, kept for generality.
    if (blockIdx.x == 0) {
        for (long long e = nChunks * CHUNK + tid; e < (long long)n; e += BLOCK) {
            int pv = pred[e];
            int tv = targ[e];
            atomicAdd(&hist[tv * NUM_CLASSES + pv], 1u);
        }
    }

    __syncthreads();

    // Merge private histogram into the global output (float atomics, L2-hot).
    for (int i = tid; i < BINS; i += BLOCK) {
        unsigned int c = hist[i];
        if (c) atomicAdd(&out[i], (float)c);  // global_atomic_add_f32
    }
}

extern "C" void kernel_launch(void* const* d_in, const int* in_sizes, int n_in,
                              void* d_out, int out_size, void* d_ws, size_t ws_size,
                              hipStream_t stream) {
    const float* cm   = (const float*)d_in[0];
    const int*   pred = (const int*)d_in[1];
    const int*   targ = (const int*)d_in[2];
    float*       out  = (float*)d_out;

    const int ncm = in_sizes[0];   // 22500
    const int n   = in_sizes[1];   // 8388608

    cm_init_kernel<<<(ncm + 255) / 256, 256, 0, stream>>>(cm, out, ncm);
    cm_hist_kernel<<<GRID, BLOCK, 0, stream>>>(pred, targ, out, n);
}